// FixedQFM_23115513987349
// MI455X (gfx1250) — compile-verified
//
#include <hip/hip_runtime.h>

// ---------------------------------------------------------------------------
// FixedQFM on MI455X (gfx1250, wave32).
//
// out[b, :] = (amp(b)[PERM]^2) @ ZVALS,  amp = kron of [cos(a*x_i/2), sin(a*x_i/2)]
//
// Memory-bound: 32 B per row -> ~1.4 us floor at 23.3 TB/s. All math in
// registers; the probs@ZVALS (B x 16)*(16 x 4) GEMM runs on
// V_WMMA_F32_16X16X4_F32 (full f32 precision), K=16 as 4 chunked WMMAs,
// with probs routed into the B-matrix layout via half-wave __shfl_xor.
// ---------------------------------------------------------------------------

typedef float v2f __attribute__((ext_vector_type(2)));
typedef float v8f __attribute__((ext_vector_type(8)));

#ifndef QFM_ALPHA
#define QFM_ALPHA 1.57f
#endif

// ---- compile-time replica of the reference _precompute() permutation ------
__host__ __device__ constexpr int perm_entry(int k) {
  // perm_final[k] = g_{i=0}(g_{i=1}(g_{i=2}(g_{i=3}(k))))  (apply i=3 first)
  int x = k;
  for (int i = 3; i >= 0; --i) {
    const int ctrl = i;
    const int tgt  = (i + 1) & 3;
    const int sc   = 3 - ctrl;
    const int st   = 3 - tgt;
    x = x ^ (((x >> sc) & 1) << st);
  }
  return x;
}
struct PermTab { int v[16]; };
__host__ __device__ constexpr PermTab make_perm() {
  PermTab t{};
  for (int k = 0; k < 16; ++k) t.v[k] = perm_entry(k);
  return t;
}
constexpr PermTab PERM_T = make_perm();

// ZVALS[k][n] = 1 - 2*bit_{3-n}(k) for n<4; padded columns are 0.
__device__ inline float zval(int k, int n) {
  return (n < 4) ? ((((k >> (3 - n)) & 1) != 0) ? -1.0f : 1.0f) : 0.0f;
}

// Per-row: permuted squared amplitudes p[0..15] (all constant-indexed -> regs)
__device__ inline void qfm_row_probs(const float4 xr, float p[16]) {
  float s0, c0, s1, c1, s2, c2, s3, c3;
  __sincosf(0.5f * QFM_ALPHA * xr.x, &s0, &c0);
  __sincosf(0.5f * QFM_ALPHA * xr.y, &s1, &c1);
  __sincosf(0.5f * QFM_ALPHA * xr.z, &s2, &c2);
  __sincosf(0.5f * QFM_ALPHA * xr.w, &s3, &c3);
  // u = kron(q0,q1), v = kron(q2,q3), amp[m] = u[m>>2] * v[m&3]
  const float u[4] = {c0 * c1, c0 * s1, s0 * c1, s0 * s1};
  const float v[4] = {c2 * c3, c2 * s3, s2 * c3, s2 * s3};
  float a[16];
#pragma unroll
  for (int m = 0; m < 16; ++m) a[m] = u[m >> 2] * v[m & 3];
#pragma unroll
  for (int k = 0; k < 16; ++k) {
    const float t = a[PERM_T.v[k]];
    p[k] = t * t;
  }
}

// ---- main kernel: 32 rows per wave, 8 waves per block (full blocks only) ---
__global__ __launch_bounds__(256) void qfm_wmma_kernel(
    const float* __restrict__ X, float* __restrict__ Out) {
  const int  lane    = threadIdx.x & 31;
  const int  waveId  = (blockIdx.x << 3) + (threadIdx.x >> 5);
  const long rowBase = (long)waveId << 5;           // 32 rows per wave
  const long row     = rowBase + lane;

  // Streaming load: one b128 per lane (coalesced).
  const float4 xr = reinterpret_cast<const float4*>(X)[row];

  float p[16];
  qfm_row_probs(xr, p);

  // A-matrix = ZVALS^T padded to 16x4 per K-chunk.
  // Documented A(16x4) f32 layout: lanes 0-15: M=lane, K=0(x)/1(y);
  //                               lanes 16-31: M=lane-16, K=2(x)/3(y).
  const int  mrow = lane & 15;                 // M index (output component)
  const int  khi  = (lane >> 4) << 1;          // lane-half K offset (0 or 2)
  const bool lo   = (lane < 16);
  v2f Az[4];
#pragma unroll
  for (int kk = 0; kk < 4; ++kk) {
    Az[kk].x = zval(4 * kk + khi + 0, mrow);
    Az[kk].y = zval(4 * kk + khi + 1, mrow);
  }

  // Two 16-row tiles per wave; B(4x16) per chunk built from own probs plus a
  // half-wave shuffle (B layout assumed to mirror A: K=0/2 split over halves).
  v8f acc0 = {0.f, 0.f, 0.f, 0.f, 0.f, 0.f, 0.f, 0.f};
  v8f acc1 = {0.f, 0.f, 0.f, 0.f, 0.f, 0.f, 0.f, 0.f};
#pragma unroll
  for (int kk = 0; kk < 4; ++kk) {
    const float p0 = p[4 * kk + 0], p1 = p[4 * kk + 1];
    const float p2 = p[4 * kk + 2], p3 = p[4 * kk + 3];
    const float x0 = __shfl_xor(p0, 16, 32);
    const float x1 = __shfl_xor(p1, 16, 32);
    const float x2 = __shfl_xor(p2, 16, 32);
    const float x3 = __shfl_xor(p3, 16, 32);
    v2f B0, B1;
    B0.x = lo ? p0 : x2;  B0.y = lo ? p1 : x3;   // tile 0: rows rowBase+0..15
    B1.x = lo ? x0 : p2;  B1.y = lo ? x1 : p3;   // tile 1: rows rowBase+16..31
    acc0 = __builtin_amdgcn_wmma_f32_16x16x4_f32(
        false, Az[kk], false, B0, (short)0, acc0, false, false);
    acc1 = __builtin_amdgcn_wmma_f32_16x16x4_f32(
        false, Az[kk], false, B1, (short)0, acc1, false, false);
  }

  // D layout: VGPR r, lanes 0-15 hold D[M=r][N=lane]. M=0..3 are the 4 output
  // components; N=lane is the row within the tile. Move tile-1 results to the
  // upper lane half so every lane emits one coalesced float4 store.
  const float t0 = __shfl_xor(acc1[0], 16, 32);
  const float t1 = __shfl_xor(acc1[1], 16, 32);
  const float t2 = __shfl_xor(acc1[2], 16, 32);
  const float t3 = __shfl_xor(acc1[3], 16, 32);
  float4 o;
  o.x = lo ? acc0[0] : t0;
  o.y = lo ? acc0[1] : t1;
  o.z = lo ? acc0[2] : t2;
  o.w = lo ? acc0[3] : t3;
  reinterpret_cast<float4*>(Out)[row] = o;
}

// ---- scalar tail for B % 256 (not taken for B = 1M; keeps EXEC==~0 in the
// WMMA kernel as the ISA requires) ------------------------------------------
__global__ void qfm_tail_kernel(const float* __restrict__ X,
                                float* __restrict__ Out,
                                int startRow, int nRows) {
  const int i = blockIdx.x * blockDim.x + threadIdx.x;
  if (i >= nRows) return;
  const long row = (long)startRow + i;
  const float4 xr = reinterpret_cast<const float4*>(X)[row];
  float p[16];
  qfm_row_probs(xr, p);
  float o[4] = {0.f, 0.f, 0.f, 0.f};
#pragma unroll
  for (int k = 0; k < 16; ++k) {
#pragma unroll
    for (int j = 0; j < 4; ++j) o[j] += p[k] * zval(k, j);
  }
  float4 ov = {o[0], o[1], o[2], o[3]};
  reinterpret_cast<float4*>(Out)[row] = ov;
}

extern "C" void kernel_launch(void* const* d_in, const int* in_sizes, int n_in,
                              void* d_out, int out_size, void* d_ws,
                              size_t ws_size, hipStream_t stream) {
  const float* X  = (const float*)d_in[0];
  float*       Out = (float*)d_out;
  const long B = (long)in_sizes[0] / 4;           // rows (BATCH)

  const long fullBlocks = B >> 8;                  // 256 rows per block
  if (fullBlocks > 0) {
    qfm_wmma_kernel<<<(unsigned)fullBlocks, 256, 0, stream>>>(X, Out);
  }
  const int rem = (int)(B - (fullBlocks << 8));
  if (rem > 0) {
    qfm_tail_kernel<<<(rem + 255) / 256, 256, 0, stream>>>(
        X, Out, (int)(fullBlocks << 8), rem);
  }
}